// cspn_3x3_naive_40338332844077
// MI455X (gfx1250) — compile-verified
//
#include <hip/hip_runtime.h>
#include <cstdint>
#include <cstddef>

namespace {

constexpr int B = 8, C = 8, H = 480, W = 640;
constexpr int N = B * H * W;
constexpr int HP = H + 2, WP = W + 2;
constexpr int NP = B * HP * WP;
constexpr float EPS = 1e-4f;
constexpr int ITER = 18;

constexpr int TW = 64, TH = 16;          // output tile per workgroup
constexpr int LW = TW + 2, LH = TH + 2;  // LDS tile with 1-px halo
constexpr int TILES_X = W / TW;          // 10
constexpr int TILES_Y = H / TH;          // 30
constexpr int NBLK = B * TILES_X * TILES_Y;  // 2400

// Channel i of guidance contributes at offset (DY[i], DX[i]) (decoded from
// the flipped one-hot edge_sum(3) taps).
constexpr int DYt[8] = {1, 1, 1, 0, 0, -1, -1, -1};
constexpr int DXt[8] = {1, 0, -1, 1, -1, 1, 0, -1};

#if __has_builtin(__builtin_amdgcn_global_load_async_to_lds_b32)
#define HAVE_ASYNC_LDS 1
#else
#define HAVE_ASYNC_LDS 0
#endif

typedef __attribute__((address_space(1))) int g_i32;
typedef __attribute__((address_space(3))) int l_i32;

__device__ __forceinline__ void wait_async0() {
#if __has_builtin(__builtin_amdgcn_s_wait_asynccnt)
  __builtin_amdgcn_s_wait_asynccnt(0);
#else
  asm volatile("s_wait_asynccnt 0" ::: "memory");
#endif
}

__global__ void k_zero(float* __restrict__ p0, float* __restrict__ p1, int n) {
  int i = blockIdx.x * blockDim.x + threadIdx.x;
  if (i < n) {
    p0[i] = 0.f;
    p1[i] = 0.f;
  }
}

// Precompute iteration-invariant AB.x = (1-blend)/aff_abs_sum and
// AB.y = (1-blend)*resid/aff_abs_sum + blend*feat_fix; seed padded feat plane.
__global__ void k_prep(const float* __restrict__ feat_init,
                       const float* __restrict__ guid,
                       const float* __restrict__ conf,
                       const float* __restrict__ ffix,
                       float2* __restrict__ AB, float* __restrict__ featPad) {
  int idx = blockIdx.x * blockDim.x + threadIdx.x;
  if (idx >= N) return;
  int x = idx % W;
  int y = (idx / W) % H;
  int b = idx / (W * H);

  float s = 0.f, sa = 0.f;
#pragma unroll
  for (int i = 0; i < 8; ++i) {
    int yy = y + DYt[i], xx = x + DXt[i];
    float g = 0.f;
    if (yy >= 0 && yy < H && xx >= 0 && xx < W)
      g = guid[((b * C + i) * H + yy) * W + xx];
    s += g;
    sa += fabsf(g);
  }
  sa += EPS;

  float fi = feat_init[idx];
  float resid = (sa - s) * fi;
  float rinv = 1.f / sa;
  float ff = ffix[idx];
  float mask = (ff > 0.f) ? 1.f : ((ff < 0.f) ? -1.f : 0.f);
  float blend = mask * conf[idx];
  float ob = 1.f - blend;

  AB[idx] = make_float2(rinv * ob, resid * rinv * ob + blend * ff);
  featPad[(b * HP + (y + 1)) * WP + (x + 1)] = fi;
}

// One propagation step. fin is a zero-padded plane; OUTPAD selects padded
// ping-pong layout vs the flat d_out layout (final iteration).
template <bool OUTPAD>
__global__ __launch_bounds__(256) void k_step(const float* __restrict__ guid,
                                              const float2* __restrict__ AB,
                                              const float* __restrict__ fin,
                                              float* __restrict__ fout) {
  __shared__ float tile[LH * LW];

  int wg = blockIdx.x;
  int tX = wg % TILES_X;
  int tY = (wg / TILES_X) % TILES_Y;
  int b = wg / (TILES_X * TILES_Y);
  int x0 = tX * TW, y0 = tY * TH;

  const float* base = fin + (size_t)b * HP * WP + (size_t)y0 * WP + x0;
  int tid = threadIdx.x;

  // Async copy the (TH+2)x(TW+2) feat halo tile global -> LDS (ASYNCcnt).
#pragma unroll
  for (int l0 = 0; l0 < LH * LW; l0 += 256) {
    int l = l0 + tid;
    if (l < LH * LW) {
      int r = l / LW;
      int c = l - r * LW;
      const float* src = base + r * WP + c;
#if HAVE_ASYNC_LDS
      __builtin_amdgcn_global_load_async_to_lds_b32((g_i32*)src,
                                                    (l_i32*)&tile[l], 0, 0);
#else
      tile[l] = src[0];
#endif
    }
  }
#if HAVE_ASYNC_LDS
  wait_async0();
#endif
  __syncthreads();

  int col = tid & (TW - 1);  // 0..63
  int rb = tid >> 6;         // 0..3
  const float* gb = guid + (size_t)b * C * H * W;
  const float2* abb = AB + (size_t)b * H * W;
  float* fo = OUTPAD ? fout + (size_t)b * HP * WP : fout + (size_t)b * H * W;

  // Per-block uniform: tiles not touching the image boundary need no clamps;
  // their gathers collapse to SGPR base + one VGPR offset + imm offsets.
  bool interior =
      (tX > 0) && (tX < TILES_X - 1) && (tY > 0) && (tY < TILES_Y - 1);

  if (interior) {
#pragma unroll
    for (int k = 0; k < 4; ++k) {
      int r = rb * 4 + k;  // 0..15
      int y = y0 + r;
      int x = x0 + col;
      int pix = y * W + x;

      float t = 0.f;
#pragma unroll
      for (int i = 0; i < 8; ++i) {
        const float* gci = gb + i * (H * W);  // uniform (SGPR) channel base
        float g = gci[pix + DYt[i] * W + DXt[i]];
        float f = tile[(r + 1 + DYt[i]) * LW + (col + 1 + DXt[i])];
        t = fmaf(g, f, t);
      }
      float2 ab = abb[pix];
      float v = fmaf(t, ab.x, ab.y);
      if (OUTPAD)
        fo[(y + 1) * WP + (x + 1)] = v;
      else
        fo[pix] = v;
    }
  } else {
#pragma unroll
    for (int k = 0; k < 4; ++k) {
      int r = rb * 4 + k;
      int y = y0 + r;
      int x = x0 + col;
      int pix = y * W + x;

      float t = 0.f;
#pragma unroll
      for (int i = 0; i < 8; ++i) {
        // OOB guidance multiplies a zero (padded) feat value, so clamping is
        // exact and branch-free.
        int yy = min(max(y + DYt[i], 0), H - 1);
        int xx = min(max(x + DXt[i], 0), W - 1);
        float g = gb[(i * H + yy) * W + xx];
        float f = tile[(r + 1 + DYt[i]) * LW + (col + 1 + DXt[i])];
        t = fmaf(g, f, t);
      }
      float2 ab = abb[pix];
      float v = fmaf(t, ab.x, ab.y);
      if (OUTPAD)
        fo[(y + 1) * WP + (x + 1)] = v;
      else
        fo[pix] = v;
    }
  }
}

}  // namespace

extern "C" void kernel_launch(void* const* d_in, const int* in_sizes, int n_in,
                              void* d_out, int out_size, void* d_ws,
                              size_t ws_size, hipStream_t stream) {
  (void)in_sizes;
  (void)n_in;
  (void)out_size;
  (void)ws_size;

  const float* feat_init = (const float*)d_in[0];
  const float* guid = (const float*)d_in[1];
  const float* conf = (const float*)d_in[2];
  const float* ffix = (const float*)d_in[3];
  float* out = (float*)d_out;

  float2* AB = (float2*)d_ws;          // N float2
  float* P0 = (float*)(AB + N);        // NP (padded ping)
  float* P1 = P0 + NP;                 // NP (padded pong)

  k_zero<<<(NP + 255) / 256, 256, 0, stream>>>(P0, P1, NP);
  k_prep<<<(N + 255) / 256, 256, 0, stream>>>(feat_init, guid, conf, ffix, AB,
                                              P0);

  float* fin = P0;
  float* fout = P1;
  for (int it = 0; it < ITER - 1; ++it) {
    k_step<true><<<NBLK, 256, 0, stream>>>(guid, AB, fin, fout);
    float* t = fin;
    fin = fout;
    fout = t;
  }
  k_step<false><<<NBLK, 256, 0, stream>>>(guid, AB, fin, out);
}